// DSA_58377195487595
// MI455X (gfx1250) — compile-verified
//
#include <hip/hip_runtime.h>
#include <hip/hip_bf16.h>

// ---------------------------------------------------------------------------
// Problem constants (fixed by setup_inputs): B=32, C=256, H=W=96
// ---------------------------------------------------------------------------
#define BB    32
#define CC    256
#define HH    96
#define WW    96
#define HWSZ  (HH * WW)          // 9216
#define NPIX  (BB * HWSZ)        // 294912 pixels
#define NBLK2 (NPIX / 256)       // 1152 blocks for the deform kernel

typedef __attribute__((ext_vector_type(2))) float v2f;
typedef __attribute__((ext_vector_type(8))) float v8f;

// ---------------------------------------------------------------------------
// K1: channel mean+max reduction.  x:(B,C,H,W) -> catmap:(B,2,H,W)
//     catmap channel 0 = max, channel 1 = mean (reference concat order).
//     float4 per thread, coalesced across hw for every c step.  ~302MB read.
// ---------------------------------------------------------------------------
__global__ void __launch_bounds__(256) k_reduce(const float* __restrict__ x,
                                                float* __restrict__ catmap) {
  const int t  = blockIdx.x * 256 + threadIdx.x;
  const int p4 = t * 4;                 // pixel index (float4 aligned; HW%4==0)
  const int b  = p4 / HWSZ;
  const int hw = p4 - b * HWSZ;
  const float* xb = x + (size_t)b * CC * HWSZ + hw;

  float4 mx = make_float4(-__FLT_MAX__, -__FLT_MAX__, -__FLT_MAX__, -__FLT_MAX__);
  float4 sm = make_float4(0.f, 0.f, 0.f, 0.f);
  for (int c = 0; c < CC; ++c) {
    __builtin_prefetch(xb + (size_t)(c + 8) * HWSZ, 0, 1);  // global_prefetch_b8
    const float4 v = *(const float4*)(xb + (size_t)c * HWSZ);
    mx.x = fmaxf(mx.x, v.x); mx.y = fmaxf(mx.y, v.y);
    mx.z = fmaxf(mx.z, v.z); mx.w = fmaxf(mx.w, v.w);
    sm.x += v.x; sm.y += v.y; sm.z += v.z; sm.w += v.w;
  }
  float* o = catmap + (size_t)b * 2 * HWSZ + hw;
  *(float4*)o = mx;
  const float inv = 1.0f / (float)CC;
  *(float4*)(o + HWSZ) = make_float4(sm.x * inv, sm.y * inv, sm.z * inv, sm.w * inv);
}

// ---------------------------------------------------------------------------
// Branchless helpers for the WMMA offset conv (18 out ch, 18 = 2*3*3 inputs)
//   W[o][i] = w_off flat (18,2,3,3);  patch index i = c*9 + ky*3 + kx
//   Clamp indices -> one unconditional load -> cndmask with the bounds mask.
//   Identical math to the reference (clip index, zero the weight) but keeps
//   the wave fully converged around the WMMA ops (no saveexec regions).
// ---------------------------------------------------------------------------
__device__ __forceinline__ float patch_val(const float* __restrict__ cmb,
                                           int y, int x, int k) {
  const int kk = k < 18 ? k : 17;       // keep address in-plane for pad lanes
  const int c  = kk / 9;
  const int r  = kk - c * 9;
  const int ky = r / 3, kx = r - ky * 3;
  const int yy = y + ky - 1, xx = x + kx - 1;
  const bool inb = (k < 18) & (yy >= 0) & (yy < HH) & (xx >= 0) & (xx < WW);
  const int yc = min(max(yy, 0), HH - 1);
  const int xc = min(max(xx, 0), WW - 1);
  const float v = cmb[c * HWSZ + yc * WW + xc];
  return inb ? v : 0.f;
}

__device__ __forceinline__ float wA_val(const float* __restrict__ w_off,
                                        int o, int k) {
  const bool inb = (o < 18) & (k < 18); // M/K padding rows/cols are zero
  const int oo = o < 18 ? o : 17;
  const int kk = k < 18 ? k : 17;
  const float v = w_off[oo * 18 + kk];
  return inb ? v : 0.f;
}

// ---------------------------------------------------------------------------
// K2 (fused): per pixel
//   (a) offset conv via V_WMMA_F32_16X16X4_F32:
//         off(18 x 16pix) = W(18x18) x patch(18x16), bias-seeded accumulator.
//       16x16x4 f32 layouts (ISA 7.12.2):
//         A 16x4 : lanes0-15 {v0=K0,v1=K1}, lanes16-31 {K2,K3}, M=lane%16
//         B 4x16 : mirrored (N=lane%16)
//         C/D    : lanes0-15 vr -> M=r ; lanes16-31 vr -> M=r+8
//       2 M-tiles x 5 K-steps = 10 WMMA per 16-pixel group; A-weights and
//       bias seeds are group-invariant and hoisted into registers.
//   (b) exchange 18 offsets through LDS, then bilinear-sample catmap at the
//       9 deformed taps, dot with w_def, write conv map + block partial sums.
// ---------------------------------------------------------------------------
__global__ void __launch_bounds__(256) k_deform(const float* __restrict__ catmap,
                                                const float* __restrict__ w_off,
                                                const float* __restrict__ b_off,
                                                const float* __restrict__ w_def,
                                                const float* __restrict__ b_def,
                                                float* __restrict__ convmap,
                                                float* __restrict__ partials) {
  __shared__ float lds_off[256][20];    // 18 used, padded stride
  __shared__ float rsum[256];
  __shared__ float rsq[256];

  const int tid  = threadIdx.x;
  const int lane = tid & 31;
  const int wv   = tid >> 5;            // wave id (8 waves)
  const int n    = lane & 15;           // matrix column / row-in-tile
  const int hs   = lane >> 4;           // half-wave select
  const int pblk = blockIdx.x * 256;    // HW%256==0 -> b uniform per block
  const int b    = pblk / HWSZ;
  const float* cmb = catmap + (size_t)b * 2 * HWSZ;

  // ---- group-invariant WMMA operands: A weights + bias seeds ----
  v2f aw0[5], aw1[5];
#pragma unroll
  for (int ks = 0; ks < 5; ++ks) {
    const int kb = ks * 4 + 2 * hs;
    aw0[ks].x = wA_val(w_off, n, kb);
    aw0[ks].y = wA_val(w_off, n, kb + 1);
    aw1[ks].x = wA_val(w_off, 16 + n, kb);
    aw1[ks].y = wA_val(w_off, 16 + n, kb + 1);
  }
  float bias0[8], bias1[8];
#pragma unroll
  for (int r = 0; r < 8; ++r) {
    bias0[r] = b_off[r + 8 * hs];                       // rows 0..15 (<18)
    const int m1 = 16 + r + 8 * hs;                     // rows 16..31
    bias1[r] = (m1 < 18) ? b_off[m1] : 0.f;
  }

  // ---- Phase A: WMMA offset conv, 2 groups of 16 pixels per wave ----
  for (int g = 0; g < 2; ++g) {
    const int pidx = wv * 32 + g * 16 + n;       // pixel-in-block this lane owns
    const int hwA  = pblk + pidx - b * HWSZ;
    const int yA   = hwA / WW;
    const int xA   = hwA - yA * WW;

    v8f acc0, acc1;
#pragma unroll
    for (int r = 0; r < 8; ++r) { acc0[r] = bias0[r]; acc1[r] = bias1[r]; }

#pragma unroll
    for (int ks = 0; ks < 5; ++ks) {             // K = 20 >= 18
      const int kb = ks * 4 + 2 * hs;
      v2f bv;                                    // B: patch values for pixel n
      bv.x = patch_val(cmb, yA, xA, kb);
      bv.y = patch_val(cmb, yA, xA, kb + 1);
      acc0 = __builtin_amdgcn_wmma_f32_16x16x4_f32(false, aw0[ks], false, bv,
                                                   (short)0, acc0, false, false);
      acc1 = __builtin_amdgcn_wmma_f32_16x16x4_f32(false, aw1[ks], false, bv,
                                                   (short)0, acc1, false, false);
    }
    // D layout -> channels: hs=0 lane holds ch 0..7 (+16,17), hs=1 holds 8..15
#pragma unroll
    for (int r = 0; r < 8; ++r) lds_off[pidx][r + 8 * hs] = acc0[r];
    if (hs == 0) { lds_off[pidx][16] = acc1[0]; lds_off[pidx][17] = acc1[1]; }
  }
  __syncthreads();

  // ---- Phase B: deformable bilinear sampling (1 pixel / thread) ----
  const int p  = pblk + tid;
  const int hw = p - b * HWSZ;
  const int y  = hw / WW;
  const int x  = hw - y * WW;
  float off[18];
#pragma unroll
  for (int i = 0; i < 18; ++i) off[i] = lds_off[tid][i];

  float conv = b_def[0];
#pragma unroll
  for (int k = 0; k < 9; ++k) {
    const int ki = k / 3, kj = k - ki * 3;
    float sy = (float)(y + ki - 1) + off[2 * k];       // off[:,k,0] = dy
    float sx = (float)(x + kj - 1) + off[2 * k + 1];   // off[:,k,1] = dx
    sy = fminf(fmaxf(sy, -1.0e6f), 1.0e6f);            // keep int cast safe
    sx = fminf(fmaxf(sx, -1.0e6f), 1.0e6f);
    const float y0f = floorf(sy), x0f = floorf(sx);
    const float ly = sy - y0f, lx = sx - x0f;
    const int y0 = (int)y0f, x0 = (int)x0f;
    float s0 = 0.f, s1 = 0.f;
#pragma unroll
    for (int cy = 0; cy < 2; ++cy) {
#pragma unroll
      for (int cx = 0; cx < 2; ++cx) {
        const int yy = y0 + cy, xx = x0 + cx;
        const float wgt = (cy ? ly : 1.f - ly) * (cx ? lx : 1.f - lx);
        const bool inb = (yy >= 0) & (yy < HH) & (xx >= 0) & (xx < WW);
        const int yc = min(max(yy, 0), HH - 1);
        const int xc = min(max(xx, 0), WW - 1);
        const float wm = inb ? wgt : 0.f;               // branchless corner
        const int idx = yc * WW + xc;
        s0 += wm * cmb[idx];                            // channel 0 = max map
        s1 += wm * cmb[HWSZ + idx];                     // channel 1 = mean map
      }
    }
    conv += w_def[k] * s0 + w_def[9 + k] * s1;  // w_def flat (1,2,3,3)
  }
  convmap[p] = conv;

  // ---- deterministic block partial sums for BN stats ----
  rsum[tid] = conv;
  rsq[tid]  = conv * conv;
  __syncthreads();
  for (int s = 128; s > 0; s >>= 1) {
    if (tid < s) { rsum[tid] += rsum[tid + s]; rsq[tid] += rsq[tid + s]; }
    __syncthreads();
  }
  if (tid == 0) {
    partials[blockIdx.x]         = rsum[0];
    partials[NBLK2 + blockIdx.x] = rsq[0];
  }
}

// ---------------------------------------------------------------------------
// K3: final BN stats (single block, fixed-order tree => deterministic).
//     stats[0] = gamma*rsqrt(var+1e-5), stats[1] = beta - mu*stats[0]
// ---------------------------------------------------------------------------
__global__ void __launch_bounds__(256) k_stats(const float* __restrict__ partials,
                                               const float* __restrict__ gamma,
                                               const float* __restrict__ beta,
                                               float* __restrict__ stats) {
  __shared__ float s1[256], s2[256];
  float a = 0.f, q = 0.f;
  for (int i = threadIdx.x; i < NBLK2; i += 256) {
    a += partials[i];
    q += partials[NBLK2 + i];
  }
  s1[threadIdx.x] = a; s2[threadIdx.x] = q;
  __syncthreads();
  for (int s = 128; s > 0; s >>= 1) {
    if (threadIdx.x < s) {
      s1[threadIdx.x] += s1[threadIdx.x + s];
      s2[threadIdx.x] += s2[threadIdx.x + s];
    }
    __syncthreads();
  }
  if (threadIdx.x == 0) {
    const float invN = 1.0f / (float)NPIX;
    const float mu  = s1[0] * invN;
    const float var = fmaxf(s2[0] * invN - mu * mu, 0.f);
    const float aa  = gamma[0] * rsqrtf(var + 1e-5f);
    stats[0] = aa;
    stats[1] = beta[0] - mu * aa;
  }
}

// ---------------------------------------------------------------------------
// K3b: conv map -> sigmoid attention map in-place (so K4 does no transcendentals
//      inside the 600MB stream; the map is reused 256x per pixel from L2).
// ---------------------------------------------------------------------------
__global__ void __launch_bounds__(256) k_sigmoid(float* __restrict__ convmap,
                                                 const float* __restrict__ stats) {
  const int p = blockIdx.x * 256 + threadIdx.x;
  const float z = convmap[p] * stats[0] + stats[1];
  convmap[p] = 1.0f / (1.0f + __expf(-z));
}

// ---------------------------------------------------------------------------
// K4: out = x * wmap  (broadcast over C).  Pure 3-stream float4 bandwidth.
// ---------------------------------------------------------------------------
__global__ void __launch_bounds__(256) k_apply(const float* __restrict__ x,
                                               const float* __restrict__ wmap,
                                               float* __restrict__ out) {
  const size_t t   = (size_t)blockIdx.x * 256 + threadIdx.x;
  const size_t e   = t * 4;              // element base (HW%4==0 => aligned)
  const size_t row = e / HWSZ;           // = b*C + c
  const int    hw  = (int)(e - row * HWSZ);
  const int    b   = (int)(row >> 8);    // C == 256
  const float4 xv = *(const float4*)(x + e);
  const float4 wv = *(const float4*)(wmap + (size_t)b * HWSZ + hw);
  float4 o;
  o.x = xv.x * wv.x; o.y = xv.y * wv.y;
  o.z = xv.z * wv.z; o.w = xv.w * wv.w;
  *(float4*)(out + e) = o;
}

// ---------------------------------------------------------------------------
extern "C" void kernel_launch(void* const* d_in, const int* in_sizes, int n_in,
                              void* d_out, int out_size, void* d_ws, size_t ws_size,
                              hipStream_t stream) {
  (void)in_sizes; (void)n_in; (void)out_size; (void)ws_size;
  const float* x     = (const float*)d_in[0];
  const float* w_off = (const float*)d_in[1];
  const float* b_off = (const float*)d_in[2];
  const float* w_def = (const float*)d_in[3];
  const float* b_def = (const float*)d_in[4];
  const float* gamma = (const float*)d_in[5];
  const float* beta  = (const float*)d_in[6];
  float* out = (float*)d_out;

  // workspace layout (floats): catmap | convmap | partials | stats  (~3.6 MB)
  float* catmap   = (float*)d_ws;
  float* convmap  = catmap + (size_t)2 * NPIX;
  float* partials = convmap + (size_t)NPIX;
  float* stats    = partials + (size_t)2 * NBLK2;

  k_reduce <<<NPIX / 1024, 256, 0, stream>>>(x, catmap);
  k_deform <<<NBLK2,       256, 0, stream>>>(catmap, w_off, b_off, w_def, b_def,
                                             convmap, partials);
  k_stats  <<<1,           256, 0, stream>>>(partials, gamma, beta, stats);
  k_sigmoid<<<NPIX / 256,  256, 0, stream>>>(convmap, stats);
  k_apply  <<<(int)(((size_t)BB * CC * HWSZ) / 1024), 256, 0, stream>>>(x, convmap, out);
}